// RGATLayer_5592047419812
// MI455X (gfx1250) — compile-verified
//
#include <hip/hip_runtime.h>
#include <hip/hip_bf16.h>
#include <math.h>

#define IN_DIM   128
#define OUT_DIM  128
#define N_RELC   5
#define ALPHA_LR 0.2f
#define NEG_INF_F (-1e30f)

typedef float v8f __attribute__((ext_vector_type(8)));
typedef float v2f __attribute__((ext_vector_type(2)));

// ---------- ordered-uint encoding for float atomic max (handles negatives) ----------
__device__ __forceinline__ unsigned encode_f(float f) {
    unsigned u = __float_as_uint(f);
    return (u & 0x80000000u) ? ~u : (u | 0x80000000u);
}
__device__ __forceinline__ float decode_f(unsigned u) {
    return (u & 0x80000000u) ? __uint_as_float(u & 0x7fffffffu) : __uint_as_float(~u);
}

// ---------- init: zero out & esum, emax = encode(-1e30) ----------
__global__ void init_kernel(float* out, unsigned* emax, float* esum, int n_out, int n_seg) {
    int i = blockIdx.x * blockDim.x + threadIdx.x;
    if (i < n_out) out[i] = 0.0f;
    if (i < n_seg) { emax[i] = encode_f(NEG_INF_F); esum[i] = 0.0f; }
}

// ---------- wqa[r][i] = sum_o WQ[r][i][o]*a_q[o] ; wka likewise with a_k ----------
__global__ void proj_a_kernel(const float* __restrict__ WQ, const float* __restrict__ WK,
                              const float* __restrict__ a_vec,
                              float* __restrict__ wqa, float* __restrict__ wka) {
    int r = blockIdx.y;
    int i = threadIdx.x;                    // 0..127
    const float* aq = a_vec;
    const float* ak = a_vec + OUT_DIM;
    const float* wq = WQ + ((size_t)r * IN_DIM + i) * OUT_DIM;
    const float* wk = WK + ((size_t)r * IN_DIM + i) * OUT_DIM;
    float sq = 0.f, sk = 0.f;
    for (int o = 0; o < OUT_DIM; ++o) { sq += wq[o] * aq[o]; sk += wk[o] * ak[o]; }
    wqa[r * IN_DIM + i] = sq;
    wka[r * IN_DIM + i] = sk;
}

// ---------- fused node scores: qs[r][n] = X[n]·wqa[r], ks[r][n] = X[n]·wka[r] ----------
// one wave (32 lanes) per node; each lane owns one float4 (128 dims / 32 lanes)
__global__ __launch_bounds__(256) void node_scores_kernel(
        const float* __restrict__ X,
        const float* __restrict__ wqa, const float* __restrict__ wka,
        float* __restrict__ qs, float* __restrict__ ks_, int nN) {
    const int lane = threadIdx.x & 31;
    const int n = blockIdx.x * (blockDim.x >> 5) + (threadIdx.x >> 5);
    if (n >= nN) return;
    const float4 x = ((const float4*)(X + (size_t)n * IN_DIM))[lane];
    float aq[N_RELC], ak[N_RELC];
#pragma unroll
    for (int r = 0; r < N_RELC; ++r) {
        const float4 wq = ((const float4*)(wqa + r * IN_DIM))[lane];
        const float4 wk = ((const float4*)(wka + r * IN_DIM))[lane];
        aq[r] = x.x * wq.x + x.y * wq.y + x.z * wq.z + x.w * wq.w;
        ak[r] = x.x * wk.x + x.y * wk.y + x.z * wk.z + x.w * wk.w;
    }
#pragma unroll
    for (int off = 16; off > 0; off >>= 1) {
#pragma unroll
        for (int r = 0; r < N_RELC; ++r) {
            aq[r] += __shfl_down(aq[r], off, 32);
            ak[r] += __shfl_down(ak[r], off, 32);
        }
    }
    if (lane == 0) {
#pragma unroll
        for (int r = 0; r < N_RELC; ++r) {
            qs[(size_t)r * nN + n] = aq[r];
            ks_[(size_t)r * nN + n] = ak[r];
        }
    }
}

// ---------- f32 WMMA GEMM with LDS-staged W: C[M x 128] = X[M x 128] @ W[128 x 128] ----------
// block = 256 threads = 8 waves; each wave: 16 rows x 128 cols (8 tiles of 16x16).
// W (64 KB) is staged once per block into LDS with coalesced float4 loads; B fragments
// then come from ds_load (16 lanes read consecutive cols at fixed k -> conflict-free banks).
// V_WMMA_F32_16X16X4_F32, K stepped by 4. M is a multiple of 16 -> per-wave uniform
// bounds check keeps EXEC all-ones inside WMMA (required by ISA).
__global__ __launch_bounds__(256) void gemm_v_kernel(const float* __restrict__ X,
                                                     const float* __restrict__ W,
                                                     float* __restrict__ C, int M) {
    __shared__ float lds_w[IN_DIM][OUT_DIM];          // 64 KB of the 320 KB WGP LDS

    // cooperative staging: 128x32 float4 chunks over 256 threads -> 16 iterations
    {
        const int tid = threadIdx.x;
#pragma unroll
        for (int it = 0; it < 16; ++it) {
            int idx = it * 256 + tid;                 // 0..4095
            int k   = idx >> 5;                       // 0..127
            int c4  = (idx & 31) << 2;                // 0,4,...,124
            float4 w = *(const float4*)(W + (size_t)k * OUT_DIM + c4);
            *(float4*)(&lds_w[k][c4]) = w;
        }
    }
    __syncthreads();

    const int wave = threadIdx.x >> 5;
    const int lane = threadIdx.x & 31;
    const int row_base = blockIdx.x * 128 + wave * 16;
    if (row_base < M) {                               // uniform per wave
        const int hl  = lane >> 4;                    // half-wave select (0/1)
        const int l16 = lane & 15;

        v8f acc[8];
#pragma unroll
        for (int t = 0; t < 8; ++t) acc[t] = (v8f)0.0f;

        for (int k = 0; k < IN_DIM; k += 4) {
            // A frag (16x4 f32): lane<16 -> K=k,k+1 ; lane>=16 -> K=k+2,k+3 ; M=l16
            const float* xr = X + (size_t)(row_base + l16) * IN_DIM + k + hl * 2;
            v2f a; a.x = xr[0]; a.y = xr[1];
            const int kk = k + hl * 2;
#pragma unroll
            for (int t = 0; t < 8; ++t) {
                // B frag (4x16 f32): v0 -> K=kk, v1 -> K=kk+1, N=l16 (from LDS)
                const int col = t * 16 + l16;
                v2f b; b.x = lds_w[kk][col]; b.y = lds_w[kk + 1][col];
                acc[t] = __builtin_amdgcn_wmma_f32_16x16x4_f32(
                    false, a, false, b, (short)0, acc[t], false, false);
            }
        }
        // C/D layout: VGPR v -> row = row_base + v + 8*hl ; col = t*16 + l16
#pragma unroll
        for (int t = 0; t < 8; ++t) {
#pragma unroll
            for (int v = 0; v < 8; ++v) {
                C[(size_t)(row_base + v + hl * 8) * OUT_DIM + t * 16 + l16] = acc[t][v];
            }
        }
    }
}

// ---------- edge pass 1: leaky-relu score + segment max ----------
__global__ void edge_max_kernel(const int* __restrict__ src, const int* __restrict__ tgt,
                                const int* __restrict__ etype,
                                const float* __restrict__ qs, const float* __restrict__ ks_,
                                float* __restrict__ edge_val, unsigned* __restrict__ emax,
                                int r, int nE, int nN) {
    int e = blockIdx.x * blockDim.x + threadIdx.x;
    if (e >= nE || etype[e] != r) return;
    int t = tgt[e], s = src[e];
    float v = qs[(size_t)r * nN + t] + ks_[(size_t)r * nN + s];
    v = (v >= 0.f) ? v : ALPHA_LR * v;
    edge_val[e] = v;
    atomicMax(&emax[(size_t)r * nN + t], encode_f(v));
}

// ---------- edge pass 2: segment sum of exp(v - max) ----------
__global__ void edge_expsum_kernel(const int* __restrict__ tgt, const int* __restrict__ etype,
                                   const float* __restrict__ edge_val,
                                   const unsigned* __restrict__ emax, float* __restrict__ esum,
                                   int r, int nE, int nN) {
    int e = blockIdx.x * blockDim.x + threadIdx.x;
    if (e >= nE || etype[e] != r) return;
    int t = tgt[e];
    float m = decode_f(emax[(size_t)r * nN + t]);
    atomicAdd(&esum[(size_t)r * nN + t], __expf(edge_val[e] - m));
}

// ---------- edge pass 3: scatter attn * Vn[src] into out[tgt] ----------
// one wave per edge; lane owns dims 4*lane..4*lane+3 (coalesced float4 gather)
__global__ __launch_bounds__(256) void edge_scatter_kernel(
        const int* __restrict__ src, const int* __restrict__ tgt,
        const int* __restrict__ etype, const float* __restrict__ edge_val,
        const unsigned* __restrict__ emax, const float* __restrict__ esum,
        const float* __restrict__ Vn, float* __restrict__ out,
        int r, int nE, int nN) {
    const int lane = threadIdx.x & 31;
    const int e = blockIdx.x * (blockDim.x >> 5) + (threadIdx.x >> 5);
    if (e >= nE || etype[e] != r) return;
    int t = tgt[e], s = src[e];
    float m   = decode_f(emax[(size_t)r * nN + t]);
    float sum = esum[(size_t)r * nN + t];
    float attn = __expf(edge_val[e] - m) / (sum + 1e-10f);
    float4 v = ((const float4*)(Vn + (size_t)s * OUT_DIM))[lane];
    float* o = out + (size_t)t * OUT_DIM + lane * 4;
    atomicAdd(&o[0], v.x * attn);
    atomicAdd(&o[1], v.y * attn);
    atomicAdd(&o[2], v.z * attn);
    atomicAdd(&o[3], v.w * attn);
}

// ---------- final ELU in place ----------
__global__ void elu_kernel(float* out, int n) {
    int i = blockIdx.x * blockDim.x + threadIdx.x;
    if (i < n) {
        float x = out[i];
        out[i] = (x > 0.f) ? x : (__expf(x) - 1.f);
    }
}

extern "C" void kernel_launch(void* const* d_in, const int* in_sizes, int n_in,
                              void* d_out, int out_size, void* d_ws, size_t ws_size,
                              hipStream_t stream) {
    const float* X      = (const float*)d_in[0];                 // [nN,128]
    const int*   eidx   = (const int*)d_in[1];                   // [2,nE]
    const int*   etype  = (const int*)d_in[2];                   // [nE]
    const float* WQ     = (const float*)d_in[3];                 // [R,128,128]
    const float* WK     = (const float*)d_in[4];
    const float* WV     = (const float*)d_in[5];
    const float* a_vec  = (const float*)d_in[6];                 // [256]
    float* out = (float*)d_out;

    const int nN = in_sizes[0] / IN_DIM;
    const int nE = in_sizes[2];
    const int nR = in_sizes[3] / (IN_DIM * OUT_DIM);
    const int* src = eidx;
    const int* tgt = eidx + nE;

    // carve workspace (256B aligned slices)
    char* wsp = (char*)d_ws;
    auto carve = [&](size_t bytes) -> void* {
        void* p = (void*)wsp;
        wsp += (bytes + 255) & ~(size_t)255;
        return p;
    };
    float*    Vn       = (float*)carve((size_t)nN * OUT_DIM * sizeof(float));
    float*    wqa      = (float*)carve((size_t)nR * IN_DIM * sizeof(float));
    float*    wka      = (float*)carve((size_t)nR * IN_DIM * sizeof(float));
    float*    qs       = (float*)carve((size_t)nR * nN * sizeof(float));
    float*    ksv      = (float*)carve((size_t)nR * nN * sizeof(float));
    unsigned* emax     = (unsigned*)carve((size_t)nR * nN * sizeof(unsigned));
    float*    esum     = (float*)carve((size_t)nR * nN * sizeof(float));
    float*    edge_val = (float*)carve((size_t)nE * sizeof(float));

    const int n_out = nN * OUT_DIM;
    const int n_seg = nR * nN;

    // 1) init accumulators
    {
        int n = n_out > n_seg ? n_out : n_seg;
        init_kernel<<<(n + 255) / 256, 256, 0, stream>>>(out, emax, esum, n_out, n_seg);
    }
    // 2) collapse Q/K GEMMs into matvecs
    proj_a_kernel<<<dim3(1, nR), IN_DIM, 0, stream>>>(WQ, WK, a_vec, wqa, wka);
    node_scores_kernel<<<(nN + 7) / 8, 256, 0, stream>>>(X, wqa, wka, qs, ksv, nN);

    // 3) per-relation: V GEMM (WMMA f32) + softmax-scatter over this relation's edges
    for (int r = 0; r < nR; ++r) {
        gemm_v_kernel<<<(nN + 127) / 128, 256, 0, stream>>>(
            X, WV + (size_t)r * IN_DIM * OUT_DIM, Vn, nN);
        edge_max_kernel<<<(nE + 255) / 256, 256, 0, stream>>>(
            src, tgt, etype, qs, ksv, edge_val, emax, r, nE, nN);
        edge_expsum_kernel<<<(nE + 255) / 256, 256, 0, stream>>>(
            tgt, etype, edge_val, emax, esum, r, nE, nN);
        edge_scatter_kernel<<<(nE + 7) / 8, 256, 0, stream>>>(
            src, tgt, etype, edge_val, emax, esum, Vn, out, r, nE, nN);
    }
    // 4) ELU
    elu_kernel<<<(n_out + 255) / 256, 256, 0, stream>>>(out, n_out);
}